// GIN_44315472560250
// MI455X (gfx1250) — compile-verified
//
#include <hip/hip_runtime.h>

#define NN 100000
#define EE 1600000
#define IND 128
#define HD 64
#define CCL 10
#define LL 4
#define GG 200
#define BN_EPS 1e-5f

typedef float v2f __attribute__((ext_vector_type(2)));
typedef float v8f __attribute__((ext_vector_type(8)));

#define SA_H 66    // LDS row stride (floats) for K=64 tiles (bank-conflict pad)
#define SA_E 130   // LDS row stride (floats) for K=128 embed tile

// ---------------------------------------------------------------------------
// Zero fill
// ---------------------------------------------------------------------------
__global__ void k_zero(float* __restrict__ p, long n) {
    long i = (long)blockIdx.x * blockDim.x + threadIdx.x;
    if (i < n) p[i] = 0.0f;
}

// ---------------------------------------------------------------------------
// Weight transpose: WT[n*K + k] = W[k*Nc + n]   (run once per launch)
// ---------------------------------------------------------------------------
__global__ void k_transpose(const float* __restrict__ W, float* __restrict__ WT,
                            int K, int Nc) {
    int t = blockIdx.x * blockDim.x + threadIdx.x;
    if (t >= K * Nc) return;
    int k = t / Nc, n = t % Nc;
    WT[n * K + k] = W[k * Nc + n];
}

// ---------------------------------------------------------------------------
// Embedding GEMM: out = X[N,128] @ W[128,64] + bias  (fp32 WMMA, LDS-staged A)
// block = 128 threads (4 waves, one 16x16 n-tile each); grid.x = N/16
// ---------------------------------------------------------------------------
__global__ __launch_bounds__(128) void k_embed(const float* __restrict__ X,
                                               const float* __restrict__ WT,
                                               const float* __restrict__ bias,
                                               float* __restrict__ out) {
    __shared__ float As[16 * SA_E];
    const int tid = threadIdx.x;
    const int mtile = blockIdx.x;
    const int m0 = mtile * 16;
    // stage A tile: 16 x 128 floats = 512 float4 chunks, 4 per thread
    #pragma unroll
    for (int c = tid; c < 512; c += 128) {
        const int row = c >> 5;              // 32 chunks per row
        const int cp = (c & 31) << 2;
        const float4 v = *(const float4*)(X + (long)(m0 + row) * IND + cp);
        float* d = &As[row * SA_E + cp];
        d[0] = v.x; d[1] = v.y; d[2] = v.z; d[3] = v.w;
    }
    __syncthreads();

    const int lane = tid & 31;
    const int wave = tid >> 5;
    const int half = lane >> 4;
    const int mr = lane & 15;
    const int n = wave * 16 + mr;
    const float* __restrict__ brow = WT + (long)n * IND;
    v8f acc = {0.f, 0.f, 0.f, 0.f, 0.f, 0.f, 0.f, 0.f};
    for (int k = 0; k < IND; k += 4) {
        const int kk = k + 2 * half;
        const float2 av = *(const float2*)(&As[mr * SA_E + kk]);
        const float2 bv = *(const float2*)(brow + kk);
        v2f a, b;
        a[0] = av.x; a[1] = av.y;
        b[0] = bv.x; b[1] = bv.y;
        acc = __builtin_amdgcn_wmma_f32_16x16x4_f32(false, a, false, b,
                                                    (short)0, acc, false, false);
    }
    const float bn = bias[n];
    #pragma unroll
    for (int r = 0; r < 8; ++r) {
        const int mm = m0 + r + 8 * half;
        out[(long)mm * HD + n] = acc[r] + bn;
    }
}

// ---------------------------------------------------------------------------
// Edge scatter: neigh[dst] += h[src]   (float4 per thread, atomic f32 adds)
// ---------------------------------------------------------------------------
__global__ __launch_bounds__(256) void k_scatter(const float* __restrict__ h,
                                                 const int* __restrict__ src,
                                                 const int* __restrict__ dst,
                                                 float* __restrict__ neigh) {
    long t = (long)blockIdx.x * blockDim.x + threadIdx.x;
    int e = (int)(t >> 4);
    if (e >= EE) return;
    int f = (int)(t & 15) << 2;
    const float4 v = *(const float4*)(h + (long)src[e] * HD + f);
    float* o = neigh + (long)dst[e] * HD + f;
    atomicAdd(o + 0, v.x);
    atomicAdd(o + 1, v.y);
    atomicAdd(o + 2, v.z);
    atomicAdd(o + 3, v.w);
}

// ---------------------------------------------------------------------------
// GEMM1: y1 = ((1+eps)*h + neigh) @ W1 + b1 ; fused BN column stats
// A transform computed once per block into LDS.
// ---------------------------------------------------------------------------
__global__ __launch_bounds__(128) void k_gemm1(const float* __restrict__ h,
                                               const float* __restrict__ neigh,
                                               const float* __restrict__ eps, int layer,
                                               const float* __restrict__ WT,
                                               const float* __restrict__ bias,
                                               float* __restrict__ y1,
                                               float* __restrict__ csum,
                                               float* __restrict__ csq) {
    __shared__ float As[16 * SA_H];
    const int tid = threadIdx.x;
    const int mtile = blockIdx.x;
    const int m0 = mtile * 16;
    const float e1 = 1.0f + eps[layer];
    // stage ((1+eps)*h + neigh) tile: 16 x 64 = 256 float4 chunks, 2 per thread
    #pragma unroll
    for (int c = tid; c < 256; c += 128) {
        const int row = c >> 4;              // 16 chunks per row
        const int cp = (c & 15) << 2;
        const long off = (long)(m0 + row) * HD + cp;
        const float4 hv = *(const float4*)(h + off);
        const float4 nv = *(const float4*)(neigh + off);
        float* d = &As[row * SA_H + cp];
        d[0] = e1 * hv.x + nv.x;
        d[1] = e1 * hv.y + nv.y;
        d[2] = e1 * hv.z + nv.z;
        d[3] = e1 * hv.w + nv.w;
    }
    __syncthreads();

    const int lane = tid & 31;
    const int wave = tid >> 5;
    const int half = lane >> 4;
    const int mr = lane & 15;
    const int n = wave * 16 + mr;
    const float* __restrict__ brow = WT + (long)n * HD;
    v8f acc = {0.f, 0.f, 0.f, 0.f, 0.f, 0.f, 0.f, 0.f};
    for (int k = 0; k < HD; k += 4) {
        const int kk = k + 2 * half;
        const float2 av = *(const float2*)(&As[mr * SA_H + kk]);
        const float2 bv = *(const float2*)(brow + kk);
        v2f a, b;
        a[0] = av.x; a[1] = av.y;
        b[0] = bv.x; b[1] = bv.y;
        acc = __builtin_amdgcn_wmma_f32_16x16x4_f32(false, a, false, b,
                                                    (short)0, acc, false, false);
    }
    const float bn = bias[n];
    float s = 0.f, ss = 0.f;
    #pragma unroll
    for (int r = 0; r < 8; ++r) {
        const int mm = m0 + r + 8 * half;
        const float v = acc[r] + bn;
        y1[(long)mm * HD + n] = v;
        s += v;
        ss += v * v;
    }
    atomicAdd(&csum[n], s);
    atomicAdd(&csq[n], ss);
}

// ---------------------------------------------------------------------------
// BN finalize: stats -> per-column (scale, shift)    (64 threads)
// ---------------------------------------------------------------------------
__global__ void k_finalize_bn(const float* __restrict__ csum,
                              const float* __restrict__ csq,
                              const float* __restrict__ gamma,
                              const float* __restrict__ beta,
                              float* __restrict__ scale,
                              float* __restrict__ shift) {
    const int c = threadIdx.x;
    if (c >= HD) return;
    const float inv_n = 1.0f / (float)NN;
    const float mean = csum[c] * inv_n;
    const float var  = csq[c] * inv_n - mean * mean;
    const float sc   = gamma[c] * rsqrtf(var + BN_EPS);
    scale[c] = sc;
    shift[c] = beta[c] - mean * sc;
}

// ---------------------------------------------------------------------------
// GEMM2: y2 = relu(scale1*y1+shift1) @ W2 + b2, row-scaled by snorm;
// fused BN2 column stats.  BN+ReLU transform staged once per block into LDS.
// ---------------------------------------------------------------------------
__global__ __launch_bounds__(128) void k_gemm2(const float* __restrict__ y1,
                                               const float* __restrict__ scale,
                                               const float* __restrict__ shift,
                                               const float* __restrict__ WT,
                                               const float* __restrict__ bias,
                                               const float* __restrict__ snorm,
                                               float* __restrict__ y2,
                                               float* __restrict__ csum,
                                               float* __restrict__ csq) {
    __shared__ float As[16 * SA_H];
    const int tid = threadIdx.x;
    const int mtile = blockIdx.x;
    const int m0 = mtile * 16;
    #pragma unroll
    for (int c = tid; c < 256; c += 128) {
        const int row = c >> 4;
        const int cp = (c & 15) << 2;
        const float4 yv = *(const float4*)(y1 + (long)(m0 + row) * HD + cp);
        const float4 sv = *(const float4*)(scale + cp);
        const float4 tv = *(const float4*)(shift + cp);
        float* d = &As[row * SA_H + cp];
        d[0] = fmaxf(sv.x * yv.x + tv.x, 0.f);
        d[1] = fmaxf(sv.y * yv.y + tv.y, 0.f);
        d[2] = fmaxf(sv.z * yv.z + tv.z, 0.f);
        d[3] = fmaxf(sv.w * yv.w + tv.w, 0.f);
    }
    __syncthreads();

    const int lane = tid & 31;
    const int wave = tid >> 5;
    const int half = lane >> 4;
    const int mr = lane & 15;
    const int n = wave * 16 + mr;
    const float* __restrict__ brow = WT + (long)n * HD;
    v8f acc = {0.f, 0.f, 0.f, 0.f, 0.f, 0.f, 0.f, 0.f};
    for (int k = 0; k < HD; k += 4) {
        const int kk = k + 2 * half;
        const float2 av = *(const float2*)(&As[mr * SA_H + kk]);
        const float2 bv = *(const float2*)(brow + kk);
        v2f a, b;
        a[0] = av.x; a[1] = av.y;
        b[0] = bv.x; b[1] = bv.y;
        acc = __builtin_amdgcn_wmma_f32_16x16x4_f32(false, a, false, b,
                                                    (short)0, acc, false, false);
    }
    const float bn = bias[n];
    float s = 0.f, ss = 0.f;
    #pragma unroll
    for (int r = 0; r < 8; ++r) {
        const int mm = m0 + r + 8 * half;
        const float v = (acc[r] + bn) * snorm[mm];
        y2[(long)mm * HD + n] = v;
        s += v;
        ss += v * v;
    }
    atomicAdd(&csum[n], s);
    atomicAdd(&csq[n], ss);
}

// ---------------------------------------------------------------------------
// Residual: h_out = h_in + relu(scale2*y2 + shift2)
// ---------------------------------------------------------------------------
__global__ __launch_bounds__(256) void k_bnres(const float* __restrict__ hin,
                                               const float* __restrict__ y2,
                                               const float* __restrict__ scale,
                                               const float* __restrict__ shift,
                                               float* __restrict__ hout) {
    long i = (long)blockIdx.x * blockDim.x + threadIdx.x;
    if (i >= (long)NN * HD) return;
    const int c = (int)(i & (HD - 1));
    const float v = scale[c] * y2[i] + shift[c];
    hout[i] = hin[i] + fmaxf(v, 0.f);
}

// ---------------------------------------------------------------------------
// Graph sum-pool: pooled[gid[node]] += hr[node]   (float4 + atomics)
// ---------------------------------------------------------------------------
__global__ __launch_bounds__(256) void k_pool(const float* __restrict__ hr,
                                              const int* __restrict__ gid,
                                              float* __restrict__ pooled) {
    long t = (long)blockIdx.x * blockDim.x + threadIdx.x;
    int node = (int)(t >> 4);
    if (node >= NN) return;
    int f = (int)(t & 15) << 2;
    const int g = gid[node];
    const float4 v = *(const float4*)(hr + (long)node * HD + f);
    float* o = pooled + (long)g * HD + f;
    atomicAdd(o + 0, v.x);
    atomicAdd(o + 1, v.y);
    atomicAdd(o + 2, v.z);
    atomicAdd(o + 3, v.w);
}

// ---------------------------------------------------------------------------
// Final scoring: out[g,c] = sum_i pooled[i] @ Wlin[i] + blin[i]
// ---------------------------------------------------------------------------
__global__ __launch_bounds__(256) void k_score(const float* __restrict__ pooled,
                                               const float* __restrict__ Wlin,
                                               const float* __restrict__ blin,
                                               float* __restrict__ out) {
    int t = blockIdx.x * blockDim.x + threadIdx.x;
    if (t >= GG * CCL) return;
    const int g = t / CCL;
    const int c = t % CCL;
    float acc = 0.f;
    for (int i = 0; i <= LL; ++i) {
        const float* p = pooled + (long)i * GG * HD + (long)g * HD;
        const float* w = Wlin + (long)i * HD * CCL + c;
        float s = 0.f;
        for (int f = 0; f < HD; ++f) s += p[f] * w[f * CCL];
        acc += s + blin[i * CCL + c];
    }
    out[t] = acc;
}

// ---------------------------------------------------------------------------
// Host orchestration
// ---------------------------------------------------------------------------
extern "C" void kernel_launch(void* const* d_in, const int* in_sizes, int n_in,
                              void* d_out, int out_size, void* d_ws, size_t ws_size,
                              hipStream_t stream) {
    const float* h_in    = (const float*)d_in[0];
    const float* snorm   = (const float*)d_in[1];
    const int*   src     = (const int*)d_in[2];
    const int*   dst     = (const int*)d_in[3];
    const int*   gid     = (const int*)d_in[4];
    const float* W_embed = (const float*)d_in[5];
    const float* b_embed = (const float*)d_in[6];
    const float* eps     = (const float*)d_in[7];
    const float* W1      = (const float*)d_in[8];
    const float* b1      = (const float*)d_in[9];
    const float* g1      = (const float*)d_in[10];
    const float* be1     = (const float*)d_in[11];
    const float* W2      = (const float*)d_in[12];
    const float* b2      = (const float*)d_in[13];
    const float* g2      = (const float*)d_in[14];
    const float* be2     = (const float*)d_in[15];
    const float* Wlin    = (const float*)d_in[16];
    const float* blin    = (const float*)d_in[17];

    float* ws = (float*)d_ws;
    const long NH = (long)NN * HD;
    float* h_cur  = ws;                 // N*H
    float* h_nxt  = ws + NH;            // N*H
    float* neigh  = ws + 2 * NH;        // N*H
    float* y1     = ws + 3 * NH;        // N*H
    float* y2     = ws + 4 * NH;        // N*H
    float* pooled = ws + 5 * NH;        // (L+1)*G*H = 64000
    float* stats  = pooled + (long)(LL + 1) * GG * HD;
    float* csum1  = stats;              // 64
    float* csq1   = stats + 64;
    float* csum2  = stats + 128;
    float* csq2   = stats + 192;
    float* scale1 = stats + 256;
    float* shift1 = stats + 320;
    float* scale2 = stats + 384;
    float* shift2 = stats + 448;
    float* wtE    = stats + 512;                       // 64*128
    float* wt1    = wtE + (long)HD * IND;              // 4 * 64*64
    float* wt2    = wt1 + (long)LL * HD * HD;          // 4 * 64*64

    const int mtiles = NN / 16;                        // 6250
    const long zn_neigh = NH;
    const int zb_neigh = (int)((zn_neigh + 255) / 256);
    const long zn_pool = (long)(LL + 1) * GG * HD;
    const int zb_pool = (int)((zn_pool + 255) / 256);
    const int scatter_blocks = (int)(((long)EE * 16 + 255) / 256);
    const int pool_blocks = (int)(((long)NN * 16 + 255) / 256);
    const int bnres_blocks = (int)((NH + 255) / 256);

    // --- one-time (per launch) weight transposes ---
    k_transpose<<<(IND * HD + 255) / 256, 256, 0, stream>>>(W_embed, wtE, IND, HD);
    for (int i = 0; i < LL; ++i) {
        k_transpose<<<(HD * HD + 255) / 256, 256, 0, stream>>>(
            W1 + (long)i * HD * HD, wt1 + (long)i * HD * HD, HD, HD);
        k_transpose<<<(HD * HD + 255) / 256, 256, 0, stream>>>(
            W2 + (long)i * HD * HD, wt2 + (long)i * HD * HD, HD, HD);
    }

    k_zero<<<zb_pool, 256, 0, stream>>>(pooled, zn_pool);

    // node embedding -> hidden_rep[0]
    k_embed<<<mtiles, 128, 0, stream>>>(h_in, wtE, b_embed, h_cur);
    k_pool<<<pool_blocks, 256, 0, stream>>>(h_cur, gid, pooled);

    for (int i = 0; i < LL; ++i) {
        k_zero<<<zb_neigh, 256, 0, stream>>>(neigh, zn_neigh);
        k_zero<<<1, 256, 0, stream>>>(stats, 256);
        k_scatter<<<scatter_blocks, 256, 0, stream>>>(h_cur, src, dst, neigh);
        k_gemm1<<<mtiles, 128, 0, stream>>>(h_cur, neigh, eps, i,
                                            wt1 + (long)i * HD * HD, b1 + i * HD,
                                            y1, csum1, csq1);
        k_finalize_bn<<<1, 64, 0, stream>>>(csum1, csq1, g1 + i * HD, be1 + i * HD,
                                            scale1, shift1);
        k_gemm2<<<mtiles, 128, 0, stream>>>(y1, scale1, shift1,
                                            wt2 + (long)i * HD * HD, b2 + i * HD,
                                            snorm, y2, csum2, csq2);
        k_finalize_bn<<<1, 64, 0, stream>>>(csum2, csq2, g2 + i * HD, be2 + i * HD,
                                            scale2, shift2);
        k_bnres<<<bnres_blocks, 256, 0, stream>>>(h_cur, y2, scale2, shift2, h_nxt);
        k_pool<<<pool_blocks, 256, 0, stream>>>(h_nxt, gid,
                                                pooled + (long)(i + 1) * GG * HD);
        float* tmp = h_cur; h_cur = h_nxt; h_nxt = tmp;
    }

    k_score<<<(GG * CCL + 255) / 256, 256, 0, stream>>>(pooled, Wlin, blin,
                                                        (float*)d_out);
}